// ROIPooler_4423816315529
// MI455X (gfx1250) — compile-verified
//
#include <hip/hip_runtime.h>
#include <hip/hip_bf16.h>
#include <cstdint>

// ROI max-pool over FPN levels for MI455X (gfx1250).
// One box per 256-thread block (8 x wave32). Channels staged 4-at-a-time into
// LDS by the Tensor Data Mover (tensor_load_to_lds, TENSORcnt double-buffered),
// 49 bin-workers x 4 channels compute the masked max from LDS.

#define ROI_MB   8      // max pixels per bin edge (reference MB)
#define CCH      4      // channels staged per TDM chunk
#define PMAX     44     // patch cap per dimension (proved <= ~42)
#define CDIM     256    // channels

typedef unsigned int  u32x4 __attribute__((ext_vector_type(4)));
typedef int           i32x4 __attribute__((ext_vector_type(4)));
typedef int           i32x8 __attribute__((ext_vector_type(8)));

#define HAVE_TDM __has_builtin(__builtin_amdgcn_tensor_load_to_lds)

__device__ __forceinline__ int clipi(int v, int hi) {
  return v < 0 ? 0 : (v > hi ? hi : v);
}
__device__ __forceinline__ int imin(int a, int b) { return a < b ? a : b; }
__device__ __forceinline__ int imax(int a, int b) { return a > b ? a : b; }

__global__ __launch_bounds__(256) void roi_max_pool_fpn(
    const float* __restrict__ p2, const float* __restrict__ p3,
    const float* __restrict__ p4, const float* __restrict__ p5,
    const float* __restrict__ boxes, float* __restrict__ out, int N)
{
  __shared__ float smem[2][CCH * PMAX * PMAX];

  const int box = blockIdx.x;
  const int tid = threadIdx.x;
  const int b   = box / N;

  // ---- box geometry (uniform across block) ----
  const float bx1 = boxes[(size_t)box * 4 + 0];
  const float by1 = boxes[(size_t)box * 4 + 1];
  const float bx2 = boxes[(size_t)box * 4 + 2];
  const float by2 = boxes[(size_t)box * 4 + 3];

  // level = clamp(floor(4 + log2(sqrt(area)/224 + 1e-8)), 2, 5) - 2
  const float area = (bx2 - bx1) * (by2 - by1);
  const float srt  = sqrtf(area);
  float lv = floorf(4.0f + log2f(srt / 224.0f + 1e-8f));
  lv = fminf(fmaxf(lv, 2.0f), 5.0f);
  const int li = (int)lv - 2;

  const float scale = 0.25f / (float)(1 << li);
  const int   H     = 200 >> li;
  const int   W     = H;
  const float* feat = (li == 0) ? p2 : (li == 1) ? p3 : (li == 2) ? p4 : p5;
  const size_t HW   = (size_t)H * (size_t)W;
  const float* fmap = feat + (size_t)b * CDIM * HW;

  // round-half-up box corners in feature coordinates
  const int x1 = (int)floorf(bx1 * scale + 0.5f);
  const int y1 = (int)floorf(by1 * scale + 0.5f);
  const int x2 = (int)floorf(bx2 * scale + 0.5f);
  const int y2 = (int)floorf(by2 * scale + 0.5f);
  const float roi_w = (float)imax(x2 - x1 + 1, 1);
  const float roi_h = (float)imax(y2 - y1 + 1, 1);
  const float bh = roi_h / 7.0f;
  const float bw = roi_w / 7.0f;

  // patch extents covering all bins (hs/he monotonic in ph)
  const int r0  = clipi(y1, H);                       // hs at ph=0
  const int c0  = clipi(x1, W);                       // ws at pw=0
  const int hs6 = clipi((int)floorf(6.0f * bh) + y1, H);
  const int he6 = clipi((int)ceilf (7.0f * bh) + y1, H);
  const int ws6 = clipi((int)floorf(6.0f * bw) + x1, W);
  const int we6 = clipi((int)ceilf (7.0f * bw) + x1, W);
  const int rowEnd = imin(he6, hs6 + ROI_MB);
  const int colEnd = imin(we6, ws6 + ROI_MB);
  const int patchH = imax(rowEnd - r0, 0);
  const int patchW = imax(colEnd - c0, 0);

  const bool useLds = (patchH > 0) && (patchW > 0) &&
                      (patchH <= PMAX) && (patchW <= PMAX);

  // ---- per-thread bin assignment: (channel-in-chunk, bin) ----
  const int c_sub  = tid >> 6;      // 0..3
  const int worker = tid & 63;      // 0..63, bins are 0..48
  int hs = 0, re = 0, ws = 0, ce = 0;
  bool empty = true;
  if (worker < 49) {
    const int ph = worker / 7;
    const int pw = worker % 7;
    hs = clipi((int)floorf((float)ph * bh) + y1, H);
    const int he = clipi((int)ceilf(((float)ph + 1.0f) * bh) + y1, H);
    ws = clipi((int)floorf((float)pw * bw) + x1, W);
    const int we = clipi((int)ceilf(((float)pw + 1.0f) * bw) + x1, W);
    empty = (he <= hs) || (we <= ws);
    re = imin(he, hs + ROI_MB);   // dynamic_slice patch clamp, exact
    ce = imin(we, ws + ROI_MB);
  }

  // ---- staging: TDM (preferred) or async-to-LDS fallback ----
  auto stage_issue = [&](int k, int buf) {
#if HAVE_TDM
    if (tid == 0) {
      const unsigned ldsAddr = (unsigned)(size_t)(void*)&smem[buf][0];
      const unsigned long long ga = (unsigned long long)(size_t)(const void*)
          (fmap + (size_t)(k * CCH) * HW + (size_t)r0 * W + (size_t)c0);
      u32x4 g0;
      g0[0] = 1u;                                     // count=1, flags 0
      g0[1] = ldsAddr;                                // lds_addr (bytes)
      g0[2] = (unsigned)(ga & 0xFFFFFFFFull);         // global_addr[31:0]
      g0[3] = (unsigned)((ga >> 32) & 0x01FFFFFFull)  // global_addr[56:32]
              | (2u << 30);                           // type=2
      i32x8 g1;
      g1[0] = 0x00020000;                             // data_size=4B, mask=0
      g1[1] = 0;                                      // tensor_dim0=1<<30 lo16
      g1[2] = 0x4000;                                 // dim0 hi16 | dim1 lo16
      g1[3] = (int)(0x4000u | ((unsigned)patchW << 16)); // dim1 hi | tile_dim0
      g1[4] = (int)((unsigned)patchH | ((unsigned)CCH << 16)); // tile1|tile2
      g1[5] = W;                                      // dim0_stride[31:0]
      g1[6] = (int)((((unsigned)HW) & 0xFFFFu) << 16);// dim1_stride lo16
      g1[7] = (int)(((unsigned)HW) >> 16);            // dim1_stride[47:16]
      i32x4 g2; g2[0] = 0x40000000; g2[1] = 0; g2[2] = 0; g2[3] = 0;
      i32x4 g3; g3[0] = 0; g3[1] = 0; g3[2] = 0; g3[3] = 0;
      i32x8 g4; g4[0] = 0; g4[1] = 0; g4[2] = 0; g4[3] = 0;
                g4[4] = 0; g4[5] = 0; g4[6] = 0; g4[7] = 0; // VADDR4: unused
      __builtin_amdgcn_tensor_load_to_lds(g0, g1, g2, g3, g4, 0);
    }
#else
    const unsigned ldsBase = (unsigned)(size_t)(void*)&smem[buf][0];
    for (int cs = 0; cs < CCH; ++cs) {
      const float* src = fmap + (size_t)(k * CCH + cs) * HW
                              + (size_t)r0 * W + (size_t)c0;
      const int wl = tid & 63;
      for (int rr = (tid >> 6); rr < patchH; rr += 4) {
        if (wl < patchW) {
          const unsigned lds = ldsBase +
              (unsigned)(((cs * patchH + rr) * patchW + wl) * 4);
          const float* g = src + (size_t)rr * W + wl;
          asm volatile("global_load_async_to_lds_b32 %0, %1, off"
                       :: "v"(lds), "v"(g) : "memory");
        }
      }
    }
#endif
  };

  auto stage_wait = [&](bool hasNext) {
#if HAVE_TDM
    if (tid == 0) {
      if (hasNext) __builtin_amdgcn_s_wait_tensorcnt(1);  // prev chunk done
      else         __builtin_amdgcn_s_wait_tensorcnt(0);
    }
#else
    (void)hasNext;
    asm volatile("s_wait_asynccnt 0x0" ::: "memory");
#endif
  };

  // ---- main loop: double-buffered chunks of CCH channels ----
  const int nChunks = CDIM / CCH;   // 64
  if (useLds) stage_issue(0, 0);

  for (int k = 0; k < nChunks; ++k) {
    const bool hasNext = (k + 1) < nChunks;
    if (useLds) {
      if (hasNext) stage_issue(k + 1, (k + 1) & 1);
      stage_wait(hasNext);
    }
    __syncthreads();

    if (worker < 49) {
      float v = 0.0f;
      if (!empty) {
        float acc = -__builtin_inff();
        if (useLds) {
          const float* plane = &smem[k & 1][(size_t)c_sub * patchH * patchW];
          for (int r = hs; r < re; ++r) {
            const float* row = plane + (size_t)(r - r0) * patchW + (ws - c0);
            for (int w = 0; w < ce - ws; ++w) acc = fmaxf(acc, row[w]);
          }
        } else {
          const float* pc = fmap + (size_t)(k * CCH + c_sub) * HW;
          for (int r = hs; r < re; ++r)
            for (int w = ws; w < ce; ++w)
              acc = fmaxf(acc, pc[(size_t)r * W + w]);
        }
        v = acc;
      }
      out[((size_t)box * CDIM + (size_t)(k * CCH + c_sub)) * 49 + worker] = v;
    }
    __syncthreads();
  }
}

extern "C" void kernel_launch(void* const* d_in, const int* in_sizes, int n_in,
                              void* d_out, int out_size, void* d_ws, size_t ws_size,
                              hipStream_t stream) {
  (void)n_in; (void)out_size; (void)d_ws; (void)ws_size;
  const float* p2    = (const float*)d_in[0];
  const float* p3    = (const float*)d_in[1];
  const float* p4    = (const float*)d_in[2];
  const float* p5    = (const float*)d_in[3];
  const float* boxes = (const float*)d_in[4];
  float* out = (float*)d_out;

  int B = in_sizes[0] / (CDIM * 200 * 200);
  if (B < 1) B = 1;
  const int nBoxes = in_sizes[4] / 4;
  const int N = nBoxes / B;

  roi_max_pool_fpn<<<nBoxes, 256, 0, stream>>>(p2, p3, p4, p5, boxes, out, N);
}